// Upsampling_88278757802288
// MI455X (gfx1250) — compile-verified
//
#include <hip/hip_runtime.h>
#include <hip/hip_bf16.h>

// ---------------------------------------------------------------------------
// Upsampling layer for MI455X (gfx1250, wave32):
//   1) kNN (k=3) brute force, p2 staged through LDS       (VALU-bound)
//   2) inverse-distance-weighted feature interpolation    (L2-resident gather)
//   3) [x1 | interp] @ W + b  via V_WMMA_F32_16X16X4_F32  (fp32 WMMA)
//   4) BatchNorm (train-mode, column reduce) + ReLU
// ---------------------------------------------------------------------------

#define N1_PTS   32768
#define N2_PTS   8192
#define D_DENSE  64
#define D_SPARSE 128
#define D_IN     192
#define D_OUT    64
#define BN_EPS   1e-5f

typedef __attribute__((ext_vector_type(2))) float v2f;
typedef __attribute__((ext_vector_type(8))) float v8f;

// ---------------------------------------------------------------------------
// Kernel 1: brute-force 3-NN + IDW weights.
// 1 thread = 1 dense point; p2 tiled through LDS (2048 pts = 24 KB / chunk).
// ---------------------------------------------------------------------------
#define KNN_BLOCK 128
#define KNN_CHUNK 2048

__global__ void __launch_bounds__(KNN_BLOCK)
knn_kernel(const float* __restrict__ p1, const float* __restrict__ p2,
           int* __restrict__ knn_idx, float* __restrict__ knn_w) {
    __shared__ float sx[KNN_CHUNK];
    __shared__ float sy[KNN_CHUNK];
    __shared__ float sz[KNN_CHUNK];

    const int n = blockIdx.x * KNN_BLOCK + threadIdx.x;   // dense point id
    const float px = p1[n * 3 + 0];
    const float py = p1[n * 3 + 1];
    const float pz = p1[n * 3 + 2];

    float d0 = 3.0e38f, d1 = 3.0e38f, d2 = 3.0e38f;
    int   i0 = 0,       i1 = 0,       i2 = 0;

    for (int base = 0; base < N2_PTS; base += KNN_CHUNK) {
        __syncthreads();
        for (int j = threadIdx.x; j < KNN_CHUNK; j += KNN_BLOCK) {
            sx[j] = p2[(base + j) * 3 + 0];
            sy[j] = p2[(base + j) * 3 + 1];
            sz[j] = p2[(base + j) * 3 + 2];
        }
        __syncthreads();

        #pragma unroll 4
        for (int j = 0; j < KNN_CHUNK; ++j) {
            const float dx = px - sx[j];
            const float dy = py - sy[j];
            const float dz = pz - sz[j];
            const float sq = fmaf(dx, dx, fmaf(dy, dy, dz * dz));
            if (sq < d2) {
                const int gj = base + j;
                if (sq < d1) {
                    d2 = d1; i2 = i1;
                    if (sq < d0) { d1 = d0; i1 = i0; d0 = sq; i0 = gj; }
                    else         { d1 = sq; i1 = gj; }
                } else {
                    d2 = sq; i2 = gj;
                }
            }
        }
    }

    // IDW weights: w_k = 1/(sqrt(d_k)+1e-8), normalized.
    const float w0 = 1.0f / (sqrtf(fmaxf(d0, 0.0f)) + 1e-8f);
    const float w1 = 1.0f / (sqrtf(fmaxf(d1, 0.0f)) + 1e-8f);
    const float w2 = 1.0f / (sqrtf(fmaxf(d2, 0.0f)) + 1e-8f);
    const float inv = 1.0f / (w0 + w1 + w2);

    knn_idx[n * 3 + 0] = i0;
    knn_idx[n * 3 + 1] = i1;
    knn_idx[n * 3 + 2] = i2;
    knn_w[n * 3 + 0] = w0 * inv;
    knn_w[n * 3 + 1] = w1 * inv;
    knn_w[n * 3 + 2] = w2 * inv;
}

// ---------------------------------------------------------------------------
// Kernel 2: interp[n, :] = sum_k w_k * x2[idx_k, :]   (128 channels, float4)
// 4 threads per point, 32 channels (8 float4) each.
// ---------------------------------------------------------------------------
__global__ void __launch_bounds__(256)
interp_kernel(const float* __restrict__ x2, const int* __restrict__ knn_idx,
              const float* __restrict__ knn_w, float* __restrict__ interp) {
    const int t = blockIdx.x * 256 + threadIdx.x;
    const int n = t >> 2;                 // point
    const int q0 = (t & 3) * 8;           // float4 slot within 32-float4 row

    const int   i0 = knn_idx[n * 3 + 0];
    const int   i1 = knn_idx[n * 3 + 1];
    const int   i2 = knn_idx[n * 3 + 2];
    const float w0 = knn_w[n * 3 + 0];
    const float w1 = knn_w[n * 3 + 1];
    const float w2 = knn_w[n * 3 + 2];

    const float4* x2v  = (const float4*)x2;      // row = 32 float4
    float4*       outv = (float4*)interp;

    #pragma unroll
    for (int q = 0; q < 8; ++q) {
        const float4 a = x2v[i0 * 32 + q0 + q];
        const float4 b = x2v[i1 * 32 + q0 + q];
        const float4 c = x2v[i2 * 32 + q0 + q];
        float4 r;
        r.x = fmaf(w0, a.x, fmaf(w1, b.x, w2 * c.x));
        r.y = fmaf(w0, a.y, fmaf(w1, b.y, w2 * c.y));
        r.z = fmaf(w0, a.z, fmaf(w1, b.z, w2 * c.z));
        r.w = fmaf(w0, a.w, fmaf(w1, b.w, w2 * c.w));
        outv[n * 32 + q0 + q] = r;
    }
}

// ---------------------------------------------------------------------------
// Kernel 3: h = [x1 | interp] @ W + b  using V_WMMA_F32_16X16X4_F32.
// One wave per 16x16 output tile; K loop 192 in steps of 4 -> 48 WMMAs/wave.
// A layout (ISA 7.12.2, 32-bit A 16x4): lanes 0-15 hold (K,K+1) of row M=lane,
// lanes 16-31 hold (K+2,K+3) of row M=lane-16.
// B layout: rows striped across lanes within a VGPR (same K split by lane half).
// C/D layout: VGPR v -> M = v (lanes 0-15) / v+8 (lanes 16-31), N = lane&15.
// ---------------------------------------------------------------------------
__device__ __forceinline__ float x_cat(const float* __restrict__ x1,
                                       const float* __restrict__ interp,
                                       int r, int k) {
    return (k < D_DENSE) ? x1[r * D_DENSE + k]
                         : interp[r * D_SPARSE + (k - D_DENSE)];
}

__global__ void __launch_bounds__(256)
gemm_wmma_kernel(const float* __restrict__ x1, const float* __restrict__ interp,
                 const float* __restrict__ W, const float* __restrict__ bias,
                 float* __restrict__ h) {
    const int wave = threadIdx.x >> 5;
    const int lane = threadIdx.x & 31;
    const int tile = blockIdx.x * 8 + wave;      // 8192 tiles total
    const int row0 = (tile >> 2) << 4;           // 2048 row tiles
    const int col0 = (tile & 3) << 4;            // 4 col tiles
    const int m     = lane & 15;
    const int khalf = (lane >> 4) << 1;          // 0 for lanes 0-15, 2 for 16-31

    v8f acc = {};

    #pragma unroll 4
    for (int k = 0; k < D_IN; k += 4) {
        const int kk = k + khalf;
        v2f a, b;
        a.x = x_cat(x1, interp, row0 + m, kk);
        a.y = x_cat(x1, interp, row0 + m, kk + 1);
        b.x = W[kk * D_OUT + col0 + m];
        b.y = W[(kk + 1) * D_OUT + col0 + m];
        acc = __builtin_amdgcn_wmma_f32_16x16x4_f32(
                  /*neg_a=*/false, a, /*neg_b=*/false, b,
                  /*c_mod=*/(short)0, acc, /*reuse_a=*/false, /*reuse_b=*/false);
    }

    const float bb   = bias[col0 + m];
    const int  rbase = row0 + ((lane >> 4) << 3);   // +8 for upper lane half
    #pragma unroll
    for (int v = 0; v < 8; ++v) {
        h[(rbase + v) * D_OUT + col0 + m] = acc[v] + bb;
    }
}

// ---------------------------------------------------------------------------
// Kernel 4a: zero the 128-float sum/sumsq region.
// Kernel 4b: per-block partial column sums (256 rows/block), float atomics out.
// ---------------------------------------------------------------------------
__global__ void zero128_kernel(float* __restrict__ p) { p[threadIdx.x] = 0.0f; }

__global__ void __launch_bounds__(64)
bn_reduce_kernel(const float* __restrict__ h, float* __restrict__ sums,
                 float* __restrict__ sumsq) {
    const int c  = threadIdx.x;            // column 0..63
    const int r0 = blockIdx.x * 256;
    float s = 0.0f, s2 = 0.0f;
    #pragma unroll 4
    for (int r = 0; r < 256; ++r) {
        const float v = h[(r0 + r) * D_OUT + c];
        s += v;
        s2 = fmaf(v, v, s2);
    }
    atomicAdd(&sums[c], s);
    atomicAdd(&sumsq[c], s2);
}

// ---------------------------------------------------------------------------
// Kernel 5: BN finalize + ReLU + assemble output tuple (p1, x, o1).
// ---------------------------------------------------------------------------
__global__ void __launch_bounds__(256)
bn_finalize_kernel(const float* __restrict__ h, const float* __restrict__ sums,
                   const float* __restrict__ sumsq, const float* __restrict__ gamma,
                   const float* __restrict__ beta, const float* __restrict__ p1,
                   const int* __restrict__ o1, float* __restrict__ out) {
    const int i = blockIdx.x * 256 + threadIdx.x;     // < N1*64
    const int c = i & 63;
    const float invN = 1.0f / (float)N1_PTS;
    const float mean = sums[c] * invN;
    const float var  = fmaf(-mean, mean, sumsq[c] * invN);
    const float rstd = rsqrtf(var + BN_EPS);
    float v = (h[i] - mean) * rstd * gamma[c] + beta[c];
    v = v > 0.0f ? v : 0.0f;
    out[N1_PTS * 3 + i] = v;

    if (i < N1_PTS * 3) out[i] = p1[i];               // tuple slot 0: p1
    if (i == 0) out[N1_PTS * 3 + N1_PTS * D_OUT] = (float)o1[0];  // slot 2: o1
}

// ---------------------------------------------------------------------------
// Host-side launch sequence.
// Inputs: p1, x1, o1, p2, x2, o2, W, b, gamma, beta
// ---------------------------------------------------------------------------
extern "C" void kernel_launch(void* const* d_in, const int* in_sizes, int n_in,
                              void* d_out, int out_size, void* d_ws, size_t ws_size,
                              hipStream_t stream) {
    const float* p1    = (const float*)d_in[0];
    const float* x1    = (const float*)d_in[1];
    const int*   o1    = (const int*)  d_in[2];
    const float* p2    = (const float*)d_in[3];
    const float* x2    = (const float*)d_in[4];
    const float* W     = (const float*)d_in[6];
    const float* bias  = (const float*)d_in[7];
    const float* gamma = (const float*)d_in[8];
    const float* beta  = (const float*)d_in[9];
    float* out = (float*)d_out;

    // Workspace layout.
    char* ws = (char*)d_ws;
    int*   knn_idx = (int*)ws;                                      // N1*3 ints
    float* knn_w   = (float*)(ws + (size_t)N1_PTS * 3 * 4);         // N1*3 floats
    float* interp  = (float*)(ws + (size_t)N1_PTS * 3 * 8);         // N1*128 floats
    float* h       = interp + (size_t)N1_PTS * D_SPARSE;            // N1*64 floats
    float* sums    = h + (size_t)N1_PTS * D_OUT;                    // 64 floats
    float* sumsq   = sums + 64;                                     // 64 floats

    knn_kernel<<<N1_PTS / KNN_BLOCK, KNN_BLOCK, 0, stream>>>(p1, p2, knn_idx, knn_w);
    interp_kernel<<<(N1_PTS * 4) / 256, 256, 0, stream>>>(x2, knn_idx, knn_w, interp);
    gemm_wmma_kernel<<<(N1_PTS / 16) * (D_OUT / 16) / 8, 256, 0, stream>>>(
        x1, interp, W, bias, h);
    zero128_kernel<<<1, 128, 0, stream>>>(sums);
    bn_reduce_kernel<<<N1_PTS / 256, 64, 0, stream>>>(h, sums, sumsq);
    bn_finalize_kernel<<<(N1_PTS * D_OUT) / 256, 256, 0, stream>>>(
        h, sums, sumsq, gamma, beta, p1, o1, out);
}